// GAE_72318659330426
// MI455X (gfx1250) — compile-verified
//
#include <hip/hip_runtime.h>

// GAE backward scan, MI455X (gfx1250, wave32).
// Memory-bound (~168 MB @ 23.3 TB/s => ~7.2 us roofline). Strategy:
//  - 1 wave per row; Tensor Data Mover DMAs the full row (rewards/values/done)
//    into LDS with hardware padding (4 dw per 16 dw) for conflict-reduced reads.
//  - Affine-map reformulation of the serial recurrence + wave32 suffix scan
//    (ds_bpermute) parallelizes the backward scan across 32 lanes.

namespace {

constexpr int T    = 2048;               // time steps (reference shape)
constexpr int CH   = 512;                // elements per wave per outer step
constexpr int PL   = 16;                 // elements per lane (CH / 32)
constexpr int PDW  = T + (T / 16) * 4;   // padded dwords per row in LDS

constexpr float GAMMA = 0.99f;
constexpr float LAM   = 0.95f;

typedef unsigned int u32;
typedef u32   u32x4 __attribute__((ext_vector_type(4)));
typedef int   i32x4 __attribute__((ext_vector_type(4)));
typedef int   i32x8 __attribute__((ext_vector_type(8)));
typedef float f32x4 __attribute__((ext_vector_type(4)));

__device__ __forceinline__ u32 lds_byte_offset(const void* p) {
  // generic -> LDS addrspace(3) -> 32-bit LDS byte offset for the TDM D#
  return (u32)(unsigned long long)(__attribute__((address_space(3))) const void*)p;
}

// One TDM descriptor: copy T dwords global->LDS, padding 4 dwords per 16 dwords.
// D# packing per CDNA5 ISA (group0 128b, group1 256b).
__device__ __forceinline__ void tdm_load_row_padded(u32 lds_byte, const void* gsrc) {
  unsigned long long ga = (unsigned long long)gsrc;
  u32x4 g0;
  g0.x = 1u;                                               // count=1, user desc
  g0.y = lds_byte;                                         // lds_addr[31:0]
  g0.z = (u32)ga;                                          // global_addr[31:0]
  g0.w = (u32)((ga >> 32) & 0x01FFFFFFull) | (2u << 30);   // addr[56:32] | type=2
  i32x8 g1;
  g1[0] = (2 << 16)      // data_size = 4B
        | (1 << 20)      // pad_enable
        | (3 << 22)      // pad_interval: every 16 dwords (64B)
        | (3 << 25);     // pad_amount: 4 dwords
  g1[1] = (int)((u32)(T & 0xFFFF) << 16);                  // tensor_dim0[15:0]
  g1[2] = (int)(((u32)T >> 16) | (1u << 16));              // dim0 hi | tensor_dim1=1
  g1[3] = (int)((u32)(T & 0xFFFF) << 16);                  // tile_dim0 = T
  g1[4] = 1;                                               // tile_dim1 = 1
  g1[5] = T;                                               // tensor_dim0_stride
  g1[6] = 0;
  g1[7] = 0;
  i32x4 gz4 = {0, 0, 0, 0};                                // groups 2/3 unused (<=2D)
  i32x8 gz8 = {0, 0, 0, 0, 0, 0, 0, 0};
  __builtin_amdgcn_tensor_load_to_lds(g0, g1, gz4, gz4, gz8, 0);
}

__device__ __forceinline__ float lane_read_f32(float x, int srcLane) {
  // wave32 gather: dst[lane] = x[srcLane]  (index in bytes)
  return __int_as_float(__builtin_amdgcn_ds_bpermute(srcLane << 2, __float_as_int(x)));
}

__global__ void __launch_bounds__(32)
gae_scan_kernel(const float* __restrict__ rewards,
                const float* __restrict__ values,
                const float* __restrict__ next_value,
                const int*   __restrict__ done_flags,
                float* __restrict__ adv_out,
                float* __restrict__ ret_out)
{
  __shared__ float sR[PDW];
  __shared__ float sV[PDW];
  __shared__ int   sD[PDW];

  const int       lane = threadIdx.x & 31;
  const int       row  = blockIdx.x;
  const long long base = (long long)row * T;

  // Async DMA: whole row of each input into LDS (TENSORcnt-tracked).
  tdm_load_row_padded(lds_byte_offset(sR), rewards    + base);
  tdm_load_row_padded(lds_byte_offset(sV), values     + base);
  tdm_load_row_padded(lds_byte_offset(sD), done_flags + base);

  const float boot = next_value[row];
  const float gl   = GAMMA * LAM;

  __builtin_amdgcn_s_wait_tensorcnt(0);
  __asm__ __volatile__("" ::: "memory");

  float carry = 0.0f;  // gae to the right of current chunk (0 beyond T)

  for (int c = T / CH - 1; c >= 0; --c) {
    const int t0 = c * CH + lane * PL;
    const int pb = t0 + ((t0 >> 4) << 2);   // padded LDS dword index (16B aligned)

    float r[PL], v[PL], dn[PL];
#pragma unroll
    for (int u = 0; u < PL / 4; ++u) {
      f32x4 rq = *(const f32x4*)(&sR[pb + 4 * u]);
      f32x4 vq = *(const f32x4*)(&sV[pb + 4 * u]);
      i32x4 dq = *(const i32x4*)(&sD[pb + 4 * u]);
#pragma unroll
      for (int k = 0; k < 4; ++k) {
        r[4 * u + k]  = rq[k];
        v[4 * u + k]  = vq[k];
        dn[4 * u + k] = (float)dq[k];
      }
    }

    // boundary element t0+16: from LDS, or bootstrap at t = T
    float vb, db;
    {
      const int w = t0 + PL;
      if (w < T) {
        const int pw = w + ((w >> 4) << 2);
        vb = sV[pw];
        db = (float)sD[pw];
      } else {
        vb = boot;          // nv[T-1] = next_value
        db = dn[PL - 1];    // nnt[T-1] = 1 - done[T-1]
      }
    }

    float delta[PL], cc[PL];
#pragma unroll
    for (int j = 0; j < PL; ++j) {
      const float nv  = (j < PL - 1) ? v[j + 1]  : vb;
      const float nd  = (j < PL - 1) ? dn[j + 1] : db;
      const float nnt = 1.0f - nd;
      delta[j] = r[j] + GAMMA * nv * nnt - v[j];
      cc[j]    = gl * nnt;
    }

    // Pass 1: fold lane segment into affine map gae_left = A + P * gae_right
    float A = 0.0f, P = 1.0f;
#pragma unroll
    for (int j = PL - 1; j >= 0; --j) { A = delta[j] + cc[j] * A; P *= cc[j]; }

    // Wave32 suffix scan of affine maps: S_l = m_l o m_{l+1} o ... o m_31
    float sA = A, sP = P;
#pragma unroll
    for (int d = 1; d < 32; d <<= 1) {
      const float a2 = lane_read_f32(sA, lane + d);
      const float p2 = lane_read_f32(sP, lane + d);
      const bool  ok = (lane + d) < 32;
      sA = ok ? (sA + sP * a2) : sA;
      sP = ok ? (sP * p2)      : sP;
    }
    // exclusive suffix (S_{l+1}); identity for lane 31
    const float eA0 = lane_read_f32(sA, lane + 1);
    const float eP0 = lane_read_f32(sP, lane + 1);
    const float eA  = (lane == 31) ? 0.0f : eA0;
    const float eP  = (lane == 31) ? 1.0f : eP0;

    // Pass 2: replay segment with the correct incoming gae; fuse returns.
    float g = eA + eP * carry;
    float adv[PL];
#pragma unroll
    for (int j = PL - 1; j >= 0; --j) { g = delta[j] + cc[j] * g; adv[j] = g; }

#pragma unroll
    for (int u = 0; u < PL / 4; ++u) {
      f32x4 aq, rq;
#pragma unroll
      for (int k = 0; k < 4; ++k) {
        aq[k] = adv[4 * u + k];
        rq[k] = adv[4 * u + k] + v[4 * u + k];
      }
      *(f32x4*)(&adv_out[base + t0 + 4 * u]) = aq;
      *(f32x4*)(&ret_out[base + t0 + 4 * u]) = rq;
    }

    // carry across chunk boundary: S_0 applied to old carry (lane 0 broadcast)
    const float cA = lane_read_f32(sA, 0);
    const float cP = lane_read_f32(sP, 0);
    carry = cA + cP * carry;
  }
}

} // anonymous namespace

extern "C" void kernel_launch(void* const* d_in, const int* in_sizes, int n_in,
                              void* d_out, int out_size, void* d_ws, size_t ws_size,
                              hipStream_t stream) {
  (void)n_in; (void)out_size; (void)d_ws; (void)ws_size;
  const float* rewards    = (const float*)d_in[0];
  const float* values     = (const float*)d_in[1];
  const float* next_value = (const float*)d_in[2];
  const int*   done_flags = (const int*)d_in[3];

  const int B = in_sizes[2];                  // next_value has B elements
  float* adv = (float*)d_out;                 // advantages [B*T]
  float* ret = adv + (size_t)in_sizes[0];     // returns    [B*T]

  gae_scan_kernel<<<B, 32, 0, stream>>>(rewards, values, next_value, done_flags,
                                        adv, ret);
}